// MemQuestionAttention_87033217286180
// MI455X (gfx1250) — compile-verified
//
#include <hip/hip_runtime.h>
#include <hip/hip_bf16.h>
#include <math.h>

// Sizes from the reference
#define BB 16
#define TT 32
#define WW 128
#define DD 1024

typedef __attribute__((ext_vector_type(2))) float v2f;
typedef __attribute__((ext_vector_type(4))) float f4;
typedef __attribute__((ext_vector_type(8))) float v8f;

// ---------------------------------------------------------------------------
// Kernel 1: q_w = source @ W_word, q_t = source @ W_turn via V_WMMA_F32_16X16X4_F32
// X: [16,1024] row-major, W: [1024,1024] row-major (y = x @ W)
// 128 waves total: wave g -> matrix (g>>6), N-tile 16*(g&63).
// A-frag (16x4 f32): lanes 0-15: K=0,1 ; lanes 16-31: K=2,3 (ISA 7.12.2)
// B-frag (4x16 f32): VGPR0 lanes0-15=K0 row, lanes16-31=K2 row; VGPR1 = K1/K3
// C/D (16x16 f32): VGPR r -> M=r (lanes 0-15), M=8+r (lanes 16-31), N=lane&15
// ---------------------------------------------------------------------------
__global__ __launch_bounds__(256) void proj_wmma(const float* __restrict__ X,
                                                 const float* __restrict__ Ww,
                                                 const float* __restrict__ Wt,
                                                 float* __restrict__ qw,
                                                 float* __restrict__ qt) {
    const int gwave = (blockIdx.x * blockDim.x + threadIdx.x) >> 5;
    const int lane  = threadIdx.x & 31;
    const int mat   = gwave >> 6;          // 0: W_word, 1: W_turn
    const int n0    = (gwave & 63) << 4;   // N tile base
    const float* __restrict__ W = mat ? Wt : Ww;
    float* __restrict__ Q = mat ? qt : qw;

    const int mrow  = lane & 15;           // M (for A) / N (for B,C)
    const int khalf = (lane >> 4) << 1;    // 0 or 2

    v8f c = {0.f, 0.f, 0.f, 0.f, 0.f, 0.f, 0.f, 0.f};
    for (int k0 = 0; k0 < DD; k0 += 4) {
        v2f a, b;
        a.x = X[mrow * DD + k0 + khalf];
        a.y = X[mrow * DD + k0 + khalf + 1];
        b.x = W[(size_t)(k0 + khalf) * DD + n0 + mrow];
        b.y = W[(size_t)(k0 + khalf + 1) * DD + n0 + mrow];
        c = __builtin_amdgcn_wmma_f32_16x16x4_f32(false, a, false, b,
                                                  (short)0, c, false, false);
    }
    const int mbase = (lane >> 4) << 3;    // 0 or 8
#pragma unroll
    for (int r = 0; r < 8; ++r) {
        Q[(size_t)(mbase + r) * DD + n0 + mrow] = c[r];
    }
}

// ---------------------------------------------------------------------------
// Kernel 2: fused word-level attention, one block per (b,t).
// Single streaming pass over memory_bank (online softmax), masked rows skipped.
// ---------------------------------------------------------------------------
__global__ __launch_bounds__(256) void word_attn_fused(const float* __restrict__ mem,
                                                       const float* __restrict__ qw,
                                                       const int* __restrict__ lens,
                                                       float* __restrict__ cw) {
    const int bt   = blockIdx.x;       // b*T + t
    const int b    = bt >> 5;
    const int wave = threadIdx.x >> 5;
    const int lane = threadIdx.x & 31;

    __shared__ float red_m[8];
    __shared__ float red_l[8];
    __shared__ float csh[DD];

    for (int i = threadIdx.x; i < DD; i += 256) csh[i] = 0.f;

    // query fragment: this lane's 32 floats of q_w[b], layout j*128 + lane*4
    f4 q[8];
    const f4* __restrict__ qv = (const f4*)(qw + (size_t)b * DD);
#pragma unroll
    for (int j = 0; j < 8; ++j) q[j] = qv[j * 32 + lane];

    const int len = lens[bt];
    const float* __restrict__ tile = mem + (size_t)bt * (WW * DD);

    float m = -INFINITY, l = 0.f;
    f4 acc[8];
#pragma unroll
    for (int j = 0; j < 8; ++j) acc[j] = (f4){0.f, 0.f, 0.f, 0.f};

    for (int w = wave; w < len; w += 8) {
        const f4* __restrict__ row = (const f4*)(tile + (size_t)w * DD);
        f4 r[8];
#pragma unroll
        for (int j = 0; j < 8; ++j)
            r[j] = __builtin_nontemporal_load(&row[j * 32 + lane]);  // stream, read-once

        float p = 0.f;
#pragma unroll
        for (int j = 0; j < 8; ++j)
            p += r[j].x * q[j].x + r[j].y * q[j].y + r[j].z * q[j].z + r[j].w * q[j].w;
#pragma unroll
        for (int off = 16; off > 0; off >>= 1) p += __shfl_xor(p, off, 32);

        const float mn    = fmaxf(m, p);
        const float scale = __expf(m - mn);   // 0 when m == -inf
        const float e     = __expf(p - mn);
        l = l * scale + e;
#pragma unroll
        for (int j = 0; j < 8; ++j) acc[j] = acc[j] * scale + r[j] * e;
        m = mn;
    }

    if (lane == 0) { red_m[wave] = m; red_l[wave] = l; }
    __syncthreads();

    float gm = -INFINITY;
#pragma unroll
    for (int i = 0; i < 8; ++i) gm = fmaxf(gm, red_m[i]);
    float gl = 0.f;
#pragma unroll
    for (int i = 0; i < 8; ++i) gl += red_l[i] * __expf(red_m[i] - gm);

    const float rs = __expf(m - gm);          // 0 for waves that saw no rows
#pragma unroll
    for (int j = 0; j < 8; ++j) {
        const int base = j * 128 + lane * 4;
        atomicAdd(&csh[base + 0], acc[j].x * rs);
        atomicAdd(&csh[base + 1], acc[j].y * rs);
        atomicAdd(&csh[base + 2], acc[j].z * rs);
        atomicAdd(&csh[base + 3], acc[j].w * rs);
    }
    __syncthreads();

    const float inv = 1.f / gl;               // len >= 1 so gl > 0
    f4* __restrict__ out = (f4*)(cw + (size_t)bt * DD);
    f4 v = ((const f4*)csh)[threadIdx.x];
    out[threadIdx.x] = v * inv;
}

// ---------------------------------------------------------------------------
// Kernel 3: turn-level attention, one block per batch b.
// ---------------------------------------------------------------------------
__global__ __launch_bounds__(256) void turn_attn(const float* __restrict__ cw,
                                                 const float* __restrict__ qt,
                                                 const int* __restrict__ turns,
                                                 float* __restrict__ out) {
    const int b    = blockIdx.x;
    const int wave = threadIdx.x >> 5;
    const int lane = threadIdx.x & 31;

    __shared__ float sc[TT];
    __shared__ float attn[TT];

    f4 q[8];
    const f4* __restrict__ qv = (const f4*)(qt + (size_t)b * DD);
#pragma unroll
    for (int j = 0; j < 8; ++j) q[j] = qv[j * 32 + lane];

    // each wave computes 4 turn scores
    for (int s = 0; s < 4; ++s) {
        const int t = wave * 4 + s;
        const f4* __restrict__ row = (const f4*)(cw + (size_t)(b * TT + t) * DD);
        float p = 0.f;
#pragma unroll
        for (int j = 0; j < 8; ++j) {
            f4 r = row[j * 32 + lane];
            p += r.x * q[j].x + r.y * q[j].y + r.z * q[j].z + r.w * q[j].w;
        }
#pragma unroll
        for (int off = 16; off > 0; off >>= 1) p += __shfl_xor(p, off, 32);
        if (lane == 0) sc[t] = p;
    }
    __syncthreads();

    // masked softmax over 32 turns, done by the first wave
    if (threadIdx.x < 32) {
        const int nt = turns[b];                  // in [1, T]
        float s = (lane < nt) ? sc[lane] : -INFINITY;
        float mx = s;
#pragma unroll
        for (int off = 16; off > 0; off >>= 1) mx = fmaxf(mx, __shfl_xor(mx, off, 32));
        const float e = __expf(s - mx);           // 0 for masked turns
        float sum = e;
#pragma unroll
        for (int off = 16; off > 0; off >>= 1) sum += __shfl_xor(sum, off, 32);
        attn[lane] = e / sum;
    }
    __syncthreads();

    // ct[b, :] = sum_t attn[t] * cw[b, t, :]
    f4 acc = {0.f, 0.f, 0.f, 0.f};
    const int d4 = threadIdx.x;                   // 256 f4 = 1024 floats
    for (int t = 0; t < TT; ++t) {
        const float a = attn[t];
        f4 v = ((const f4*)(cw + (size_t)(b * TT + t) * DD))[d4];
        acc += v * a;
    }
    ((f4*)(out + (size_t)b * DD))[d4] = acc;
}

// ---------------------------------------------------------------------------
extern "C" void kernel_launch(void* const* d_in, const int* in_sizes, int n_in,
                              void* d_out, int out_size, void* d_ws, size_t ws_size,
                              hipStream_t stream) {
    const float* source = (const float*)d_in[0];   // [16,1,1024]
    const float* mem    = (const float*)d_in[1];   // [16,32,128,1024]
    const int*   lens   = (const int*)d_in[2];     // [16,32]
    const int*   turns  = (const int*)d_in[3];     // [16]
    const float* Ww     = (const float*)d_in[4];   // [1024,1024]
    const float* Wt     = (const float*)d_in[5];   // [1024,1024]
    float* out = (float*)d_out;                    // [16,1024]

    float* qw = (float*)d_ws;                      // 16*1024
    float* qt = qw + BB * DD;                      // 16*1024
    float* cw = qt + BB * DD;                      // 16*32*1024

    proj_wmma<<<16, 256, 0, stream>>>(source, Ww, Wt, qw, qt);
    word_attn_fused<<<BB * TT, 256, 0, stream>>>(mem, qw, lens, cw);
    turn_attn<<<BB, 256, 0, stream>>>(cw, qt, turns, out);
}